// PointFeatureNet_12884901888411
// MI455X (gfx1250) — compile-verified
//
#include <hip/hip_runtime.h>
#include <math.h>

#define EPS_F   1e-5f
#define SLOPE_F 0.2f

static constexpr int Bn  = 8;
static constexpr int Nn  = 2048;
static constexpr int Kn  = 20;
static constexpr int BN  = Bn * Nn;      // 16384
static constexpr int J20 = BN * Kn;      // 327680  (columns for 4D convs)
static constexpr int J1  = BN;           // 16384   (columns for 1D convs)

typedef float v2f __attribute__((ext_vector_type(2)));
typedef float v8f __attribute__((ext_vector_type(8)));
typedef unsigned int u32x4 __attribute__((ext_vector_type(4)));
typedef int i32x4 __attribute__((ext_vector_type(4)));
typedef int i32x8 __attribute__((ext_vector_type(8)));

__device__ __forceinline__ float lrelu_f(float x) { return x >= 0.f ? x : SLOPE_F * x; }

// ---------------------------------------------------------------- transpose in
// pointcloud (B,N,3) -> XYZ0 (3,B,N)  channel-major
__global__ void k_transpose_in(const float* __restrict__ pc, float* __restrict__ xyz) {
  int i = blockIdx.x * blockDim.x + threadIdx.x;   // over B*N
  if (i >= BN) return;
  #pragma unroll
  for (int c = 0; c < 3; ++c) xyz[(size_t)c * BN + i] = pc[(size_t)i * 3 + c];
}

// ---------------------------------------------------------------- kNN (top-k of pd)
// X layout (C,B,N). pd = 2*inner - |q|^2 - |m|^2 ; sorted descending; write ids[kk*dsel].
template <int C>
__global__ void k_knn(const float* __restrict__ X, int kfull, int dsel,
                      int* __restrict__ idx_out) {
  constexpr int TILE = 192;
  __shared__ float ts[TILE * C];
  __shared__ float xxm[TILE];
  const int b = blockIdx.y;
  const int n = blockIdx.x * blockDim.x + threadIdx.x;
  const float* Xb = X + (size_t)b * Nn;

  float q[C];
  float xxq = 0.f;
  #pragma unroll
  for (int c = 0; c < C; ++c) { q[c] = Xb[(size_t)c * BN + n]; xxq += q[c] * q[c]; }

  float vals[80]; int ids[80]; int cnt = 0;

  for (int m0 = 0; m0 < Nn; m0 += TILE) {
    const int mt = min(TILE, Nn - m0);
    __syncthreads();
    for (int i = threadIdx.x; i < mt * C; i += blockDim.x) {
      int mm = i / C, c = i % C;
      ts[mm * C + c] = Xb[(size_t)c * BN + m0 + mm];
    }
    __syncthreads();
    for (int mm = threadIdx.x; mm < mt; mm += blockDim.x) {
      float s = 0.f;
      #pragma unroll
      for (int c = 0; c < C; ++c) { float v = ts[mm * C + c]; s += v * v; }
      xxm[mm] = s;
    }
    __syncthreads();
    for (int mm = 0; mm < mt; ++mm) {
      float inner = 0.f;
      #pragma unroll
      for (int c = 0; c < C; ++c) inner += q[c] * ts[mm * C + c];
      float pd = 2.f * inner - xxq - xxm[mm];
      if (cnt == kfull && pd <= vals[kfull - 1]) continue;
      int pos = (cnt < kfull) ? cnt : (kfull - 1);
      while (pos > 0 && vals[pos - 1] < pd) {
        vals[pos] = vals[pos - 1]; ids[pos] = ids[pos - 1]; --pos;
      }
      vals[pos] = pd; ids[pos] = m0 + mm;
      if (cnt < kfull) ++cnt;
    }
  }
  int* op = idx_out + ((size_t)b * Nn + n) * Kn;
  for (int kk = 0; kk < Kn; ++kk) op[kk] = ids[kk * dsel];
}

// ---------------------------------------------------------------- edge feature gather
// X (C,B,N), idx (B,N,Kn) -> F (Cpad2, B,N,Kn): [feat-center ; center ; zero pad]
__global__ void k_gather(const float* __restrict__ X, const int* __restrict__ idx,
                         int C, int Cpad2, float* __restrict__ F) {
  int j = blockIdx.x * blockDim.x + threadIdx.x;   // over J20
  if (j >= J20) return;
  int bn = j / Kn;
  int b  = bn / Nn;
  int n  = bn - b * Nn;
  int nb = idx[j];
  const float* Xb = X + (size_t)b * Nn;
  for (int c = 0; c < C; ++c) {
    float xc = Xb[(size_t)c * BN + n];
    float xn = Xb[(size_t)c * BN + nb];
    F[(size_t)c * J20 + j]       = xn - xc;
    F[(size_t)(C + c) * J20 + j] = xc;
  }
  for (int c = 2 * C; c < Cpad2; ++c) F[(size_t)c * J20 + j] = 0.f;
}

// ---------------------------------------------------------------- WMMA f32 GEMM
// Y(O x J) = W(O x Creal) * F(Cpad x J) + bias.
// Block = 4 waves; wave computes two 16x16 tiles (cols j0, j0+16) sharing the A frag.
// Weight tile (16 rows x Cpad cols, zero-filled past Creal) staged in LDS by the
// Tensor Data Mover (TDM OOB reads return zero -> free Cpad padding).
__launch_bounds__(128)
__global__ void k_gemm_wmma(const float* __restrict__ W, const float* __restrict__ F,
                            const float* __restrict__ bias, float* __restrict__ Y,
                            int Creal, int Cpad, int J) {
  extern __shared__ float Wlds[];          // 16 * Cpad floats
  const int o0 = blockIdx.y * 16;

#if __has_builtin(__builtin_amdgcn_tensor_load_to_lds)
  if (threadIdx.x == 0) {                  // wave 0 issues the DMA once
    unsigned long long ga =
        (unsigned long long)(const void*)(W + (size_t)o0 * Creal);
    u32x4 g0;
    g0[0] = 1u;                                        // count=1 (valid user D#)
    g0[1] = (unsigned)__builtin_amdgcn_groupstaticsize(); // lds_addr = dynamic LDS base
    g0[2] = (unsigned)ga;                              // global_addr[31:0]
    g0[3] = (unsigned)((ga >> 32) & 0x01FFFFFFull) | (2u << 30); // addr[56:32]|type=2
    i32x8 g1;
    g1[0] = (2 << 16);                                 // data_size=4B, mask=0
    g1[1] = (int)(((unsigned)Creal & 0xFFFFu) << 16);  // tensor_dim0 lo16
    g1[2] = (int)(((unsigned)Creal >> 16) | (16u << 16)); // td0 hi16 | tensor_dim1=16 lo
    g1[3] = (int)(((unsigned)Cpad & 0xFFFFu) << 16);   // td1 hi (0) | tile_dim0=Cpad
    g1[4] = 16;                                        // tile_dim1=16, tile_dim2=0
    g1[5] = Creal;                                     // tensor_dim0_stride lo32
    g1[6] = 0;                                         // stride0 hi | stride1 lo
    g1[7] = 0;
    i32x4 z4 = (i32x4){0, 0, 0, 0};
    i32x8 z8 = (i32x8){0, 0, 0, 0, 0, 0, 0, 0};
    __builtin_amdgcn_tensor_load_to_lds(g0, g1, z4, z4, z8, 0);
    __builtin_amdgcn_s_wait_tensorcnt(0);
  }
  __syncthreads();
#else
  for (int i = threadIdx.x; i < 16 * Cpad; i += blockDim.x) {
    int r = i / Cpad, c = i - r * Cpad;
    Wlds[i] = (c < Creal) ? W[(size_t)(o0 + r) * Creal + c] : 0.f;
  }
  __syncthreads();
#endif

  const int wave = threadIdx.x >> 5;       // wave32
  const int lane = threadIdx.x & 31;
  const int half = lane >> 4;              // 0: rows K0/K1, 1: rows K2/K3
  const int l15  = lane & 15;
  const int j0   = blockIdx.x * 128 + wave * 32 + l15;

  union { v8f v; float f[8]; } acc0, acc1;
  acc0.v = (v8f){0.f, 0.f, 0.f, 0.f, 0.f, 0.f, 0.f, 0.f};
  acc1.v = acc0.v;

  const float* p = F + j0 + (size_t)(2 * half) * J;  // strength-reduced B pointer
  int aoff = l15 * Cpad + 2 * half;

  for (int c0 = 0; c0 < Cpad; c0 += 4) {
    v2f a;  a.x  = Wlds[aoff];  a.y  = Wlds[aoff + 1];
    v2f b0; b0.x = p[0];        b0.y = p[J];
    v2f b1; b1.x = p[16];       b1.y = p[J + 16];
    __builtin_prefetch(p + (size_t)8 * J, 0, 1);     // unconditional, no branch
    acc0.v = __builtin_amdgcn_wmma_f32_16x16x4_f32(false, a, false, b0,
                                                   (short)0, acc0.v, false, false);
    acc1.v = __builtin_amdgcn_wmma_f32_16x16x4_f32(false, a, false, b1,
                                                   (short)0, acc1.v, false, false);
    aoff += 4;
    p += (size_t)4 * J;
  }
  #pragma unroll
  for (int r = 0; r < 8; ++r) {
    int o = o0 + r + 8 * half;               // C/D layout: M = r (+8 for upper half)
    float bo = bias[o];
    Y[(size_t)o * J + j0]      = acc0.f[r] + bo;
    Y[(size_t)o * J + j0 + 16] = acc1.f[r] + bo;
  }
}

// ---------------------------------------------------------------- per-channel sum/sumsq
__global__ void k_stats(const float* __restrict__ Y, int J, float* __restrict__ st) {
  __shared__ float s1[256], s2[256];
  const int o = blockIdx.y;
  const int base = blockIdx.x * 8192;
  const float* yo = Y + (size_t)o * J;
  float a = 0.f, b = 0.f;
  for (int i = threadIdx.x; i < 8192 && base + i < J; i += 256) {
    float v = yo[base + i]; a += v; b += v * v;
  }
  s1[threadIdx.x] = a; s2[threadIdx.x] = b;
  __syncthreads();
  for (int w = 128; w > 0; w >>= 1) {
    if (threadIdx.x < w) { s1[threadIdx.x] += s1[threadIdx.x + w];
                           s2[threadIdx.x] += s2[threadIdx.x + w]; }
    __syncthreads();
  }
  if (threadIdx.x == 0) { atomicAdd(&st[2 * o], s1[0]); atomicAdd(&st[2 * o + 1], s2[0]); }
}

// ---------------------------------------------------------------- BN + lrelu (elementwise)
__global__ void k_norm_lrelu(const float* __restrict__ Y, const float* __restrict__ st,
                             const float* __restrict__ g, const float* __restrict__ beta,
                             int J, float* __restrict__ Z, long long total) {
  long long i = (long long)blockIdx.x * blockDim.x + threadIdx.x;
  if (i >= total) return;
  int o = (int)(i / J);
  float invJ = 1.f / (float)J;
  float m = st[2 * o] * invJ;
  float v = st[2 * o + 1] * invJ - m * m;
  float z = (Y[i] - m) * rsqrtf(v + EPS_F) * g[o] + beta[o];
  Z[i] = lrelu_f(z);
}

// ---------------------------------------------------------------- BN + lrelu + max_k (+res, +output)
// Y (O,B,N,Kn) -> outw (O,B,N); optional residual (O,B,N); optional d_out (B,O,N)
__global__ void k_norm_lrelu_maxk(const float* __restrict__ Y, const float* __restrict__ st,
                                  const float* __restrict__ g, const float* __restrict__ beta,
                                  int O, const float* __restrict__ res,
                                  float* __restrict__ outw, float* __restrict__ outd) {
  int i = blockIdx.x * blockDim.x + threadIdx.x;   // over O*BN
  if (i >= O * BN) return;
  int o = i / BN, bn = i - o * BN;
  float invJ = 1.f / (float)J20;
  float m = st[2 * o] * invJ;
  float v = st[2 * o + 1] * invJ - m * m;
  float sc = rsqrtf(v + EPS_F) * g[o];
  const float* yp = Y + (size_t)o * J20 + (size_t)bn * Kn;
  float mx = -INFINITY;
  #pragma unroll
  for (int k = 0; k < Kn; ++k) mx = fmaxf(mx, lrelu_f((yp[k] - m) * sc + beta[o]));
  if (res) mx += res[i];
  outw[i] = mx;
  if (outd) {
    int b = bn / Nn, n = bn - b * Nn;
    outd[(size_t)b * O * Nn + (size_t)o * Nn + n] = mx;
  }
}

// ---------------------------------------------------------------- max over N : (O,B,N)->(B,O)
__global__ void k_max_n(const float* __restrict__ X, int O, float* __restrict__ out) {
  int i = blockIdx.x * blockDim.x + threadIdx.x;   // over O*Bn
  if (i >= O * Bn) return;
  int o = i / Bn, b = i - o * Bn;
  const float* xp = X + (size_t)o * BN + b * Nn;
  float mx = -INFINITY;
  for (int n = 0; n < Nn; ++n) mx = fmaxf(mx, xp[n]);
  out[b * O + o] = mx;
}

// ---------------------------------------------------------------- dense + BN(batch) + lrelu
__global__ void k_dense_bn_lrelu(const float* __restrict__ Xin, const float* __restrict__ W,
                                 const float* __restrict__ bias, const float* __restrict__ g,
                                 const float* __restrict__ beta, int IN, int OUT,
                                 float* __restrict__ Out) {
  int o = blockIdx.x * blockDim.x + threadIdx.x;
  if (o >= OUT) return;
  float y[Bn];
  #pragma unroll
  for (int b = 0; b < Bn; ++b) y[b] = bias[o];
  for (int i = 0; i < IN; ++i) {
    float w = W[(size_t)o * IN + i];
    #pragma unroll
    for (int b = 0; b < Bn; ++b) y[b] += w * Xin[(size_t)b * IN + i];
  }
  float m = 0.f;
  #pragma unroll
  for (int b = 0; b < Bn; ++b) m += y[b];
  m *= (1.f / Bn);
  float v = 0.f;
  #pragma unroll
  for (int b = 0; b < Bn; ++b) { float d = y[b] - m; v += d * d; }
  v *= (1.f / Bn);
  float sc = rsqrtf(v + EPS_F) * g[o];
  #pragma unroll
  for (int b = 0; b < Bn; ++b) Out[(size_t)b * OUT + o] = lrelu_f((y[b] - m) * sc + beta[o]);
}

// ---------------------------------------------------------------- quaternion head
__global__ void k_trans(const float* __restrict__ L2v, const float* __restrict__ Wt,
                        const float* __restrict__ bt, float* __restrict__ TRws,
                        float* __restrict__ outd) {
  int b = threadIdx.x;
  if (b >= Bn) return;
  float q[4];
  for (int o = 0; o < 4; ++o) {
    float s = bt[o] + (o == 0 ? 1.f : 0.f);
    for (int i = 0; i < 256; ++i) s += Wt[o * 256 + i] * L2v[b * 256 + i];
    q[o] = s;
  }
  float ss = 2.f / (q[0]*q[0] + q[1]*q[1] + q[2]*q[2] + q[3]*q[3]);
  float R[9];
  R[0] = 1.f - ss*(q[2]*q[2] + q[3]*q[3]); R[1] = ss*(q[1]*q[2] - q[0]*q[3]); R[2] = ss*(q[1]*q[3] + q[0]*q[2]);
  R[3] = ss*(q[1]*q[2] + q[0]*q[3]); R[4] = 1.f - ss*(q[1]*q[1] + q[3]*q[3]); R[5] = ss*(q[2]*q[3] - q[0]*q[1]);
  R[6] = ss*(q[1]*q[3] - q[0]*q[2]); R[7] = ss*(q[2]*q[3] + q[0]*q[1]); R[8] = 1.f - ss*(q[1]*q[1] + q[2]*q[2]);
  for (int r = 0; r < 9; ++r) { TRws[b * 9 + r] = R[r]; outd[b * 9 + r] = R[r]; }
}

// ---------------------------------------------------------------- rotate: bdn = sum_c bcn * bcd
__global__ void k_rotate(const float* __restrict__ X0, const float* __restrict__ TR,
                         float* __restrict__ XR, float* __restrict__ outxyz) {
  int i = blockIdx.x * blockDim.x + threadIdx.x;   // over BN
  if (i >= BN) return;
  int b = i / Nn, n = i - b * Nn;
  float x = X0[0 * BN + i], y = X0[1 * BN + i], z = X0[2 * BN + i];
  const float* T = TR + b * 9;
  #pragma unroll
  for (int d = 0; d < 3; ++d) {
    float v = x * T[0 * 3 + d] + y * T[1 * 3 + d] + z * T[2 * 3 + d];
    XR[(size_t)d * BN + i] = v;
    outxyz[(size_t)b * 3 * Nn + (size_t)d * Nn + n] = v;
  }
}

// ---------------------------------------------------------------- concat [s3 ; broadcast(max_n s3)]
__global__ void k_concat(const float* __restrict__ S3Zp, const float* __restrict__ S3Mp,
                         float* __restrict__ SC) {
  int i = blockIdx.x * blockDim.x + threadIdx.x;   // over 512*J1
  if (i >= 512 * J1) return;
  int c = i / J1, r = i - c * J1;
  if (c < 256) SC[i] = S3Zp[(size_t)c * J1 + r];
  else { int b = r / Nn; SC[i] = S3Mp[b * 256 + (c - 256)]; }
}

// ---------------------------------------------------------------- deconv3 (3x64, bias only)
__global__ void k_offsets(const float* __restrict__ D2Zp, const float* __restrict__ W3,
                          const float* __restrict__ b3, float* __restrict__ outd) {
  int i = blockIdx.x * blockDim.x + threadIdx.x;   // over BN
  if (i >= BN) return;
  int b = i / Nn, n = i - b * Nn;
  #pragma unroll
  for (int o = 0; o < 3; ++o) {
    float s = b3[o];
    for (int c = 0; c < 64; ++c) s += W3[o * 64 + c] * D2Zp[(size_t)c * J1 + i];
    outd[(size_t)b * 3 * Nn + (size_t)o * Nn + n] = s;
  }
}

// ================================================================ host
extern "C" void kernel_launch(void* const* d_in, const int* in_sizes, int n_in,
                              void* d_out, int out_size, void* d_ws, size_t ws_size,
                              hipStream_t stream) {
  (void)in_sizes; (void)n_in; (void)out_size; (void)ws_size;
  const float* PC = (const float*)d_in[0];
  auto P = [&](int i) { return (const float*)d_in[i]; };
  // d_in order (setup_inputs dict order, layers store w,b[,g,beta]):
  // 0 pointcloud | pfn c1:1-4 c2:5-8 c3:9-12 c4:13-16
  // tnet c1:17-20 c2:21-24 c3:25-28 lin1:29-32 lin2:33-36 trans:37-38
  // snet c1:39-42 c2:43-46 c3:47-50 d1:51-54 d2:55-58 d3:59-60

  float* ws = (float*)d_ws;
  size_t cur = 0;
  auto alloc = [&](size_t nfl) { float* p = ws + cur; cur += (nfl + 63) & ~(size_t)63; return p; };
  float* XYZ0    = alloc((size_t)3 * BN);
  float* XYZR    = alloc((size_t)3 * BN);
  int*   IDX     = (int*)alloc((size_t)J20);
  float* FEAT6   = alloc((size_t)8 * J20);
  float* FEAT128 = alloc((size_t)128 * J20);
  float* YBUF    = alloc((size_t)128 * J20);
  float* ZBUF    = alloc((size_t)64 * J20);   // also holds (1024,J1)
  float* T2M     = alloc((size_t)128 * J1);
  float* S2M     = alloc((size_t)128 * J1);
  float* S3Z     = alloc((size_t)256 * J1);
  float* S3M     = alloc((size_t)Bn * 256);
  float* SCAT    = alloc((size_t)512 * J1);
  float* D1Z     = alloc((size_t)128 * J1);
  float* D2Z     = alloc((size_t)64 * J1);
  float* X1      = alloc((size_t)64 * J1);
  float* X2      = alloc((size_t)64 * J1);
  float* X3      = alloc((size_t)64 * J1);
  float* X4      = alloc((size_t)64 * J1);
  float* T3M     = alloc((size_t)Bn * 1024);
  float* L1      = alloc((size_t)Bn * 512);
  float* L2      = alloc((size_t)Bn * 256);
  float* TR      = alloc((size_t)Bn * 9);
  float* ST      = alloc((size_t)2 * 1024);

  float* OUT     = (float*)d_out;
  float* out_x2  = OUT;
  float* out_x3  = out_x2 + (size_t)Bn * 64 * Nn;
  float* out_x4  = out_x3 + (size_t)Bn * 64 * Nn;
  float* out_tr  = out_x4 + (size_t)Bn * 64 * Nn;
  float* out_off = out_tr + (size_t)Bn * 9;
  float* out_xyz = out_off + (size_t)Bn * 3 * Nn;

  auto gemm = [&](const float* W, const float* F, const float* bias, float* Y,
                  int Creal, int Cpad, int O, int J) {
    dim3 g(J / 128, O / 16);
    k_gemm_wmma<<<g, 128, 16 * Cpad * sizeof(float), stream>>>(W, F, bias, Y, Creal, Cpad, J);
  };
  auto stats = [&](const float* Y, int O, int J) {
    (void)hipMemsetAsync(ST, 0, 2 * O * sizeof(float), stream);
    dim3 g((J + 8191) / 8192, O);
    k_stats<<<g, 256, 0, stream>>>(Y, J, ST);
  };
  auto norm = [&](const float* Y, const float* g_, const float* be, int O, int J, float* Z) {
    long long total = (long long)O * J;
    k_norm_lrelu<<<(unsigned)((total + 255) / 256), 256, 0, stream>>>(Y, ST, g_, be, J, Z, total);
  };
  auto normmaxk = [&](const float* Y, const float* g_, const float* be, int O,
                      const float* res, float* ow, float* od) {
    int tot = O * BN;
    k_norm_lrelu_maxk<<<(tot + 255) / 256, 256, 0, stream>>>(Y, ST, g_, be, O, res, ow, od);
  };

  // -------- stage 0: xyz0, knn, edge features
  k_transpose_in<<<BN / 256, 256, 0, stream>>>(PC, XYZ0);
  k_knn<3><<<dim3(Nn / 128, Bn), 128, 0, stream>>>(XYZ0, 20, 1, IDX);
  k_gather<<<J20 / 256, 256, 0, stream>>>(XYZ0, IDX, 3, 8, FEAT6);

  // -------- transform net
  gemm(P(17), FEAT6, P(18), YBUF, 6, 8, 64, J20);
  stats(YBUF, 64, J20);  norm(YBUF, P(19), P(20), 64, J20, ZBUF);
  gemm(P(21), ZBUF, P(22), YBUF, 64, 64, 128, J20);
  stats(YBUF, 128, J20); normmaxk(YBUF, P(23), P(24), 128, nullptr, T2M, nullptr);
  gemm(P(25), T2M, P(26), YBUF, 128, 128, 1024, J1);
  stats(YBUF, 1024, J1); norm(YBUF, P(27), P(28), 1024, J1, ZBUF);
  k_max_n<<<(1024 * Bn + 255) / 256, 256, 0, stream>>>(ZBUF, 1024, T3M);
  k_dense_bn_lrelu<<<2, 256, 0, stream>>>(T3M, P(29), P(30), P(31), P(32), 1024, 512, L1);
  k_dense_bn_lrelu<<<1, 256, 0, stream>>>(L1, P(33), P(34), P(35), P(36), 512, 256, L2);
  k_trans<<<1, 32, 0, stream>>>(L2, P(37), P(38), TR, out_tr);
  k_rotate<<<BN / 256, 256, 0, stream>>>(XYZ0, TR, XYZR, out_xyz);

  // -------- knn + edge features on rotated cloud (shared by snet & pfn.conv1)
  k_knn<3><<<dim3(Nn / 128, Bn), 128, 0, stream>>>(XYZR, 20, 1, IDX);
  k_gather<<<J20 / 256, 256, 0, stream>>>(XYZR, IDX, 3, 8, FEAT6);

  // -------- shift net
  gemm(P(39), FEAT6, P(40), YBUF, 6, 8, 64, J20);
  stats(YBUF, 64, J20);  norm(YBUF, P(41), P(42), 64, J20, ZBUF);
  gemm(P(43), ZBUF, P(44), YBUF, 64, 64, 128, J20);
  stats(YBUF, 128, J20); normmaxk(YBUF, P(45), P(46), 128, nullptr, S2M, nullptr);
  gemm(P(47), S2M, P(48), YBUF, 128, 128, 256, J1);
  stats(YBUF, 256, J1);  norm(YBUF, P(49), P(50), 256, J1, S3Z);
  k_max_n<<<(256 * Bn + 255) / 256, 256, 0, stream>>>(S3Z, 256, S3M);
  k_concat<<<(512 * J1) / 256, 256, 0, stream>>>(S3Z, S3M, SCAT);
  gemm(P(51), SCAT, P(52), YBUF, 512, 512, 128, J1);
  stats(YBUF, 128, J1);  norm(YBUF, P(53), P(54), 128, J1, D1Z);
  gemm(P(55), D1Z, P(56), YBUF, 128, 128, 64, J1);
  stats(YBUF, 64, J1);   norm(YBUF, P(57), P(58), 64, J1, D2Z);
  k_offsets<<<BN / 256, 256, 0, stream>>>(D2Z, P(59), P(60), out_off);

  // -------- point feature net
  gemm(P(1), FEAT6, P(2), YBUF, 6, 8, 64, J20);
  stats(YBUF, 64, J20); normmaxk(YBUF, P(3), P(4), 64, nullptr, X1, nullptr);

  k_knn<64><<<dim3(Nn / 128, Bn), 128, 0, stream>>>(X1, 80, 4, IDX);
  k_gather<<<J20 / 256, 256, 0, stream>>>(X1, IDX, 64, 128, FEAT128);
  gemm(P(5), FEAT128, P(6), YBUF, 128, 128, 64, J20);
  stats(YBUF, 64, J20); normmaxk(YBUF, P(7), P(8), 64, X1, X2, out_x2);

  k_knn<64><<<dim3(Nn / 128, Bn), 128, 0, stream>>>(X2, 20, 1, IDX);
  k_gather<<<J20 / 256, 256, 0, stream>>>(X2, IDX, 64, 128, FEAT128);
  gemm(P(9), FEAT128, P(10), YBUF, 128, 128, 64, J20);
  stats(YBUF, 64, J20); normmaxk(YBUF, P(11), P(12), 64, X2, X3, out_x3);

  k_knn<64><<<dim3(Nn / 128, Bn), 128, 0, stream>>>(X3, 80, 4, IDX);
  k_gather<<<J20 / 256, 256, 0, stream>>>(X3, IDX, 64, 128, FEAT128);
  gemm(P(13), FEAT128, P(14), YBUF, 128, 128, 64, J20);
  stats(YBUF, 64, J20); normmaxk(YBUF, P(15), P(16), 64, X3, X4, out_x4);
}